// CustomMinkowskiConvolutionTranspose_8177617732129
// MI455X (gfx1250) — compile-verified
//
#include <hip/hip_runtime.h>

typedef __attribute__((ext_vector_type(2))) float v2f;
typedef __attribute__((ext_vector_type(8))) float v8f;

#define CIN  32
#define COUT 32
#define WAVES_PER_BLOCK 8
#define ROWS_PER_WAVE   16

__global__ __launch_bounds__(256) void zero_out_kernel(float* __restrict__ out, int n) {
    int i = blockIdx.x * blockDim.x + threadIdx.x;
    int stride = gridDim.x * blockDim.x;
    for (; i < n; i += stride) out[i] = 0.0f;
}

__global__ __launch_bounds__(256) void spconvT_wmma_kernel(
    const float* __restrict__ feats,      // [N_IN, 32]
    const float* __restrict__ kernel,     // [K, 32, 32]
    const int*   __restrict__ in_map,     // [K, M]
    const int*   __restrict__ out_map,    // [K, M]
    float*       __restrict__ out,        // [N_OUT, 32]
    int M)
{
    const int k      = blockIdx.y;
    const int wave   = threadIdx.x >> 5;
    const int lane   = threadIdx.x & 31;
    const int lanelo = lane & 15;
    const int hi     = lane >> 4;          // 0 for lanes 0-15, 1 for lanes 16-31

    const int m0 = (blockIdx.x * WAVES_PER_BLOCK + wave) * ROWS_PER_WAVE;
    if (m0 >= M) return;

    // ---- Preload B fragments for this kernel offset: 8 K-chunks x 2 N-tiles ----
    // B (4x16 f32, 2 VGPRs): VGPR0 = K rows {0 (lo lanes), 2 (hi lanes)},
    //                        VGPR1 = K rows {1, 3};  N = lane % 16 (+16 for tile 1)
    const float* __restrict__ Wk = kernel + (long)k * (CIN * COUT);
    v2f Bfrag[8][2];
#pragma unroll
    for (int c = 0; c < 8; ++c) {
        const int krow = 4 * c + 2 * hi;
#pragma unroll
        for (int t = 0; t < 2; ++t) {
            const int col = 16 * t + lanelo;
            v2f b;
            b.x = Wk[(krow + 0) * COUT + col];
            b.y = Wk[(krow + 1) * COUT + col];
            Bfrag[c][t] = b;
        }
    }

    // ---- Gather row index for A fragments (same row for both lane halves) ----
    int m  = m0 + lanelo;
    int mc = (m < M) ? m : (M - 1);                 // clamp; EXEC must stay all-1s for WMMA
    const int row = in_map[(long)k * M + mc];
    const float* __restrict__ frow = feats + (long)row * CIN;

    // ---- Accumulate D[16x32] via 8 x (2 WMMA f32 16x16x4) ----
    v8f acc0 = {};
    v8f acc1 = {};
#pragma unroll
    for (int c = 0; c < 8; ++c) {
        // A (16x4 f32, 2 VGPRs): VGPR0 = K {0 (lo), 2 (hi)}, VGPR1 = K {1, 3}
        // -> lane loads feats[row][4c + 2*hi + {0,1}] as one aligned float2
        v2f a = *(const v2f*)(frow + 4 * c + 2 * hi);
        acc0 = __builtin_amdgcn_wmma_f32_16x16x4_f32(
            false, a, false, Bfrag[c][0], (short)0, acc0, false, false);
        acc1 = __builtin_amdgcn_wmma_f32_16x16x4_f32(
            false, a, false, Bfrag[c][1], (short)0, acc1, false, false);
    }

    // ---- Scatter-add: D VGPR v, lane -> element (M = v + 8*hi, N = lanelo + 16*t) ----
#pragma unroll
    for (int v = 0; v < 8; ++v) {
        const int mm = m0 + v + 8 * hi;
        if (mm < M) {
            const int orow = out_map[(long)k * M + mm];
            float* obase = out + (long)orow * COUT + lanelo;
            atomicAdd(obase,      acc0[v]);
            atomicAdd(obase + 16, acc1[v]);
        }
    }
}

extern "C" void kernel_launch(void* const* d_in, const int* in_sizes, int n_in,
                              void* d_out, int out_size, void* d_ws, size_t ws_size,
                              hipStream_t stream) {
    const float* feats   = (const float*)d_in[0];
    const float* kern    = (const float*)d_in[1];
    const int*   in_map  = (const int*)d_in[2];
    const int*   out_map = (const int*)d_in[3];
    float*       out     = (float*)d_out;

    // Derive K and M from input sizes (C_IN = C_OUT = 32 per reference).
    const int K = in_sizes[1] / (CIN * COUT);
    const int M = in_sizes[2] / K;

    // Zero-initialize the output accumulator (harness poisons d_out).
    {
        int blocks = (out_size + 255) / 256;
        if (blocks > 4096) blocks = 4096;   // grid-stride
        zero_out_kernel<<<blocks, 256, 0, stream>>>(out, out_size);
    }

    dim3 grid((M + WAVES_PER_BLOCK * ROWS_PER_WAVE - 1) / (WAVES_PER_BLOCK * ROWS_PER_WAVE), K);
    spconvT_wmma_kernel<<<grid, 256, 0, stream>>>(feats, kern, in_map, out_map, out, M);
}